// Breadth_59545426592366
// MI455X (gfx1250) — compile-verified
//
#include <hip/hip_runtime.h>
#include <math.h>

typedef __attribute__((ext_vector_type(2))) float v2f;
typedef __attribute__((ext_vector_type(8))) float v8f;

#define N_NODES   50000
#define N_EDGES   640000
#define E_TOT     (N_EDGES + N_NODES)   // self-loops appended
#define DIM       128
#define NEG_SLOPE 0.2f

// ---------------------------------------------------------------------------
// 1) init: zero output accumulator, -inf segment max, zero denom
// ---------------------------------------------------------------------------
__global__ void k_init(float* __restrict__ out, float* __restrict__ maxv,
                       float* __restrict__ denom) {
    int i = blockIdx.x * blockDim.x + threadIdx.x;   // 6.4M threads exactly
    out[i] = 0.0f;
    if (i < N_NODES) {
        maxv[i]  = -__builtin_inff();
        denom[i] = 0.0f;
    }
}

// ---------------------------------------------------------------------------
// 2) H = X @ W via V_WMMA_F32_16X16X4_F32 (wave32, 16x16 tile per wave)
//    block = 256 threads (8 waves): blockIdx.x -> M tile (16 rows),
//    wave id -> N tile (16 cols). 50000 = 3125*16, no guards, EXEC all-1s.
// ---------------------------------------------------------------------------
__global__ void k_gemm_wmma(const float* __restrict__ X,
                            const float* __restrict__ W,
                            float* __restrict__ H) {
    const int wave = threadIdx.x >> 5;
    const int lane = threadIdx.x & 31;
    const int hb   = lane >> 4;      // lane half: selects K pair (A) / K row pair (B)
    const int l    = lane & 15;      // M row (A) / N col (B,C,D)
    const int m0   = blockIdx.x * 16;
    const int n0   = wave * 16;

    // A fragment source: row m0+l, cols {k+2*hb, k+2*hb+1}  (float2, 8B aligned)
    const float* xrow = X + (m0 + l) * DIM + 2 * hb;
    // B fragment source: col n0+l, rows {k+2*hb, k+2*hb+1}
    const float* wcol = W + (2 * hb) * DIM + n0 + l;

    v8f acc = {};
#pragma unroll
    for (int k = 0; k < DIM; k += 4) {
        v2f a = *(const v2f*)(xrow + k);
        v2f b;
        b.x = wcol[k * DIM];
        b.y = wcol[k * DIM + DIM];
        acc = __builtin_amdgcn_wmma_f32_16x16x4_f32(
            false, a, false, b, (short)0, acc, false, false);
    }

    // D layout: lanes 0-15 -> rows 0..7 in VGPR 0..7; lanes 16-31 -> rows 8..15
    float* hout = H + (m0 + 8 * hb) * DIM + n0 + l;
#pragma unroll
    for (int r = 0; r < 8; ++r)
        hout[r * DIM] = acc[r];
}

// ---------------------------------------------------------------------------
// 3) per-node attention coefficients: a_s = h . att_src, a_d = h . att_dst
//    one wave32 per node, float4 per lane, shfl_xor reduction
// ---------------------------------------------------------------------------
__global__ void k_attn_coef(const float* __restrict__ H,
                            const float* __restrict__ att_src,
                            const float* __restrict__ att_dst,
                            float* __restrict__ a_s, float* __restrict__ a_d) {
    const int wave = threadIdx.x >> 5;
    const int lane = threadIdx.x & 31;
    const int node = blockIdx.x * 8 + wave;
    if (node >= N_NODES) return;

    float4 h4  = ((const float4*)(H + node * DIM))[lane];
    float4 as4 = ((const float4*)att_src)[lane];
    float4 ad4 = ((const float4*)att_dst)[lane];
    float s = h4.x * as4.x + h4.y * as4.y + h4.z * as4.z + h4.w * as4.w;
    float d = h4.x * ad4.x + h4.y * ad4.y + h4.z * ad4.z + h4.w * ad4.w;
#pragma unroll
    for (int off = 16; off > 0; off >>= 1) {
        s += __shfl_xor(s, off, 32);
        d += __shfl_xor(d, off, 32);
    }
    if (lane == 0) { a_s[node] = s; a_d[node] = d; }
}

// ---------------------------------------------------------------------------
// helpers
// ---------------------------------------------------------------------------
__device__ __forceinline__ void edge_st(const int* __restrict__ ei, int e,
                                        int& s, int& t) {
    if (e < N_EDGES) { s = ei[e]; t = ei[N_EDGES + e]; }
    else             { s = t = e - N_EDGES; }          // self-loop
}

__device__ __forceinline__ float leaky(float e) {
    return e > 0.0f ? e : NEG_SLOPE * e;
}

// exact float atomic max via monotone int reinterpretation (native int atomics)
__device__ __forceinline__ void atomicMaxF32(float* addr, float v) {
    if (v >= 0.0f) atomicMax((int*)addr, __float_as_int(v));
    else           atomicMin((unsigned int*)addr, __float_as_uint(v));
}

// ---------------------------------------------------------------------------
// 4) segment max of LeakyReLU logits over dst
// ---------------------------------------------------------------------------
__global__ void k_edge_max(const int* __restrict__ ei,
                           const float* __restrict__ a_s,
                           const float* __restrict__ a_d,
                           float* __restrict__ maxv) {
    int e = blockIdx.x * blockDim.x + threadIdx.x;
    if (e >= E_TOT) return;
    int s, t; edge_st(ei, e, s, t);
    atomicMaxF32(&maxv[t], leaky(a_s[s] + a_d[t]));
}

// ---------------------------------------------------------------------------
// 5) segment sum of exp(e - max[dst])
// ---------------------------------------------------------------------------
__global__ void k_edge_denom(const int* __restrict__ ei,
                             const float* __restrict__ a_s,
                             const float* __restrict__ a_d,
                             const float* __restrict__ maxv,
                             float* __restrict__ denom) {
    int e = blockIdx.x * blockDim.x + threadIdx.x;
    if (e >= E_TOT) return;
    int s, t; edge_st(ei, e, s, t);
    float w = expf(leaky(a_s[s] + a_d[t]) - maxv[t]);
    unsafeAtomicAdd(&denom[t], w);
}

// ---------------------------------------------------------------------------
// 6) weighted scatter-add: out[dst] += alpha * h[src]; one wave per edge,
//    float4 gather per lane + 4 native f32 atomics per lane
// ---------------------------------------------------------------------------
__global__ void k_scatter(const int* __restrict__ ei,
                          const float* __restrict__ H,
                          const float* __restrict__ a_s,
                          const float* __restrict__ a_d,
                          const float* __restrict__ maxv,
                          const float* __restrict__ denom,
                          float* __restrict__ out) {
    const int wave = threadIdx.x >> 5;
    const int lane = threadIdx.x & 31;
    const int e = blockIdx.x * 8 + wave;
    if (e >= E_TOT) return;
    int s, t; edge_st(ei, e, s, t);

    float alpha = expf(leaky(a_s[s] + a_d[t]) - maxv[t]) / denom[t];

    float4 h4 = ((const float4*)(H + s * DIM))[lane];
    float* o = out + t * DIM + lane * 4;
    unsafeAtomicAdd(o + 0, alpha * h4.x);
    unsafeAtomicAdd(o + 1, alpha * h4.y);
    unsafeAtomicAdd(o + 2, alpha * h4.z);
    unsafeAtomicAdd(o + 3, alpha * h4.w);
}

// ---------------------------------------------------------------------------
// 7) out = tanh(acc + bias)
// ---------------------------------------------------------------------------
__global__ void k_finalize(float* __restrict__ out,
                           const float* __restrict__ bias) {
    int i = blockIdx.x * blockDim.x + threadIdx.x;   // 6.4M threads exactly
    out[i] = tanhf(out[i] + bias[i & (DIM - 1)]);
}

// ---------------------------------------------------------------------------
// launcher
// ---------------------------------------------------------------------------
extern "C" void kernel_launch(void* const* d_in, const int* in_sizes, int n_in,
                              void* d_out, int out_size, void* d_ws, size_t ws_size,
                              hipStream_t stream) {
    const float* x        = (const float*)d_in[0];
    const int*   ei       = (const int*)  d_in[1];   // [2, N_EDGES]
    const float* W        = (const float*)d_in[2];
    const float* att_src  = (const float*)d_in[3];
    const float* att_dst  = (const float*)d_in[4];
    const float* bias     = (const float*)d_in[5];
    float*       out      = (float*)d_out;

    // workspace layout (floats)
    float* H     = (float*)d_ws;                    // 50000*128
    float* a_s   = H    + (size_t)N_NODES * DIM;    // 50000
    float* a_d   = a_s  + N_NODES;                  // 50000
    float* maxv  = a_d  + N_NODES;                  // 50000
    float* denom = maxv + N_NODES;                  // 50000

    const int NE_ELEMS = N_NODES * DIM;             // 6,400,000

    k_init<<<NE_ELEMS / 256, 256, 0, stream>>>(out, maxv, denom);

    k_gemm_wmma<<<N_NODES / 16, 256, 0, stream>>>(x, W, H);

    k_attn_coef<<<N_NODES / 8, 256, 0, stream>>>(H, att_src, att_dst, a_s, a_d);

    const int eb = (E_TOT + 255) / 256;
    k_edge_max<<<eb, 256, 0, stream>>>(ei, a_s, a_d, maxv);
    k_edge_denom<<<eb, 256, 0, stream>>>(ei, a_s, a_d, maxv, denom);

    k_scatter<<<(E_TOT + 7) / 8, 256, 0, stream>>>(ei, H, a_s, a_d, maxv, denom, out);

    k_finalize<<<NE_ELEMS / 256, 256, 0, stream>>>(out, bias);
}